// MPNN_64854006170309
// MI455X (gfx1250) — compile-verified
//
#include <hip/hip_runtime.h>

// ---------------------------------------------------------------------------
// MPNN (MetaLayer x3) forward for MI455X (gfx1250, wave32).
// - Dense per-edge/per-node MLPs via V_WMMA_F32_16X16X4_F32 (fp32 end-to-end).
// - One wave per 16-row tile; concatenated input rows staged in LDS with
//   GLOBAL_LOAD_ASYNC_TO_LDS_B128 (ASYNCcnt) -- no VGPR round-trip.
// - Weights pre-packed fragment-major so each WMMA B operand is a single
//   coalesced global_load_b64 per lane.
// ---------------------------------------------------------------------------

#define N_NODES_C  20000
#define N_EDGES_C  320000
#define N_GRAPHS_C 128
#define AT_STRIDE  260   // K_max = 256 (+4 pad); 1040 B row stride (16B align)

typedef float v2f __attribute__((ext_vector_type(2)));
typedef float v8f __attribute__((ext_vector_type(8)));

static __device__ __forceinline__ v8f wmma_f32_16x16x4(v2f a, v2f b, v8f c) {
  // 8 args: (neg_a, A, neg_b, B, c_mod, C, reuse_a, reuse_b)
  return __builtin_amdgcn_wmma_f32_16x16x4_f32(false, a, false, b, (short)0, c,
                                               false, false);
}

static __device__ __forceinline__ void atomicAddF(float* p, float v) {
  __hip_atomic_fetch_add(p, v, __ATOMIC_RELAXED, __HIP_MEMORY_SCOPE_AGENT);
}

// Low 32 bits of the flat address of an LDS object = LDS byte offset
// (flat LDS aperture).
static __device__ __forceinline__ unsigned lds_off_of(const void* p) {
  return (unsigned)(unsigned long long)p;
}

// 16B memory -> LDS async copy (tracked by ASYNCcnt, no VGPR data).
static __device__ __forceinline__ void async_copy_b128(unsigned lds_off,
                                                       const float* gaddr) {
  asm volatile("global_load_async_to_lds_b128 %0, %1, off"
               :
               : "v"(lds_off), "v"(gaddr)
               : "memory");
}

static __device__ __forceinline__ void wait_async() {
  asm volatile("s_wait_asynccnt 0x0" ::: "memory");
}

// Gather 16 rows of width W (floats, W % 4 == 0) from p[idx[r]] into the LDS
// tile at column `off` using async b128 copies. Uniform trip count (no EXEC
// divergence): 16*W % 128 == 0 for all W used here.
template <int W>
static __device__ __forceinline__ void gather_rows(const float* __restrict__ p,
                                                   const int* idx,
                                                   float (*At)[AT_STRIDE],
                                                   int off, int lane) {
  for (int i = lane * 4; i < 16 * W; i += 128) {
    int r = i / W, c = i % W;
    async_copy_b128(lds_off_of(&At[r][off + c]),
                    p + (size_t)idx[r] * W + c);
  }
}

// Gather 16 consecutive rows starting at row0 (async b128).
template <int W>
static __device__ __forceinline__ void gather_block(const float* __restrict__ p,
                                                    int row0,
                                                    float (*At)[AT_STRIDE],
                                                    int off, int lane) {
  for (int i = lane * 4; i < 16 * W; i += 128) {
    int r = i / W, c = i % W;
    async_copy_b128(lds_off_of(&At[r][off + c]),
                    p + (size_t)(row0 + r) * W + c);
  }
}

template <int NCH>
static __device__ __forceinline__ void zero_acc(v8f (&acc)[NCH]) {
#pragma unroll
  for (int c = 0; c < NCH; ++c)
    acc[c] = (v8f){0.f, 0.f, 0.f, 0.f, 0.f, 0.f, 0.f, 0.f};
}

// 16xK (LDS tile) x KxDOUT (Wf, fragment-major) -> 16xDOUT in NCH accums.
// A frag (32b A 16x4): lanes0-15 rows M=0..15 hold K=k0,k0+1; lanes16-31 hold
// K=k0+2,k0+3.  B frags are pre-packed in lane order: one b64 load per WMMA.
template <int K, int NCH>
static __device__ __forceinline__ void gemm_tile(const float (*At)[AT_STRIDE],
                                                 const float* __restrict__ Wf,
                                                 v8f (&acc)[NCH], int lane) {
  const int m = lane & 15;
  const int koff = (lane >> 4) << 1;  // 0 for lanes 0-15, 2 for lanes 16-31
  const float* __restrict__ w = Wf + lane * 2;
  for (int k0 = 0; k0 < K; k0 += 4) {
    v2f a;
    a.x = At[m][k0 + koff];
    a.y = At[m][k0 + koff + 1];
#pragma unroll
    for (int c = 0; c < NCH; ++c) {
      v2f b = *(const v2f*)(w + ((k0 >> 2) * NCH + c) * 64);
      acc[c] = wmma_f32_16x16x4(a, b, acc[c]);
    }
  }
}

// ---------------------------------------------------------------------------
// Edge kernel: e = act(W_e @ [x[src],x[dst],e_old,u[bsrc]])
//              msg = act(W_n1 @ [x[dst],x[src],e,u[bdst]]) -> atomic into agg
//              e also atomically accumulated into edge_sum[batch[dst]]
// One wave per 16-edge tile.  In-place e (reads own rows before writing).
// ---------------------------------------------------------------------------
template <int DN, int DE, int DG, int EO, int NO, bool RELU>
__global__ void __launch_bounds__(32) mpnn_edge_kernel(
    const float* __restrict__ x, const float* __restrict__ ein,
    const float* __restrict__ u, const int* __restrict__ src,
    const int* __restrict__ dst, const int* __restrict__ batch,
    const float* __restrict__ WfE, const float* __restrict__ ebias,
    const float* __restrict__ WfN1, const float* __restrict__ n1bias,
    float* __restrict__ eout, float* __restrict__ agg,
    float* __restrict__ edge_sum) {
  __shared__ float At[16][AT_STRIDE];
  __shared__ int sS[16], sD[16], sBs[16], sBd[16];
  const int lane = threadIdx.x;
  const int e0 = blockIdx.x * 16;

  if (lane < 16) {
    int s = src[e0 + lane], d = dst[e0 + lane];
    sS[lane] = s;
    sD[lane] = d;
    sBs[lane] = batch[s];
    sBd[lane] = batch[d];
  }
  __syncthreads();

  // A1 = [x[src] | x[dst] | e_old | u[batch[src]]]   (async -> LDS)
  gather_rows<DN>(x, sS, At, 0, lane);
  gather_rows<DN>(x, sD, At, DN, lane);
  gather_block<DE>(ein, e0, At, 2 * DN, lane);
  gather_rows<DG>(u, sBs, At, 2 * DN + DE, lane);
  wait_async();
  __syncthreads();

  constexpr int KE = 2 * DN + DE + DG;
  constexpr int NCH = EO / 16;
  v8f acc[NCH];
  zero_acc(acc);
  gemm_tile<KE, NCH>(At, WfE, acc, lane);

  const int hi = lane >> 4, nn = lane & 15;
  float ev[NCH][8];
#pragma unroll
  for (int c = 0; c < NCH; ++c)
#pragma unroll
    for (int r = 0; r < 8; ++r) {
      float v = acc[c][r] + ebias[c * 16 + nn];
      if (RELU) v = fmaxf(v, 0.f);
      ev[c][r] = v;
    }

  __syncthreads();
  // A2 = [x[dst] | x[src] | e_new | u[batch[dst]]]
  gather_rows<DN>(x, sD, At, 0, lane);
  gather_rows<DN>(x, sS, At, DN, lane);
  gather_rows<DG>(u, sBd, At, 2 * DN + EO, lane);
#pragma unroll
  for (int c = 0; c < NCH; ++c)
#pragma unroll
    for (int r = 0; r < 8; ++r) {
      int mrow = r + 8 * hi;  // D layout: VGPR r, high half-wave -> M=r+8
      float v = ev[c][r];
      At[mrow][2 * DN + c * 16 + nn] = v;
      eout[(size_t)(e0 + mrow) * EO + c * 16 + nn] = v;
      atomicAddF(&edge_sum[(size_t)sBd[mrow] * EO + c * 16 + nn], v);
    }
  wait_async();
  __syncthreads();

  constexpr int K1 = 2 * DN + EO + DG;
  constexpr int NCH2 = NO / 16;
  v8f acc2[NCH2];
  zero_acc(acc2);
  gemm_tile<K1, NCH2>(At, WfN1, acc2, lane);
#pragma unroll
  for (int c = 0; c < NCH2; ++c)
#pragma unroll
    for (int r = 0; r < 8; ++r) {
      int mrow = r + 8 * hi;
      float v = acc2[c][r] + n1bias[c * 16 + nn];
      if (RELU) v = fmaxf(v, 0.f);
      atomicAddF(&agg[(size_t)sD[mrow] * NO + c * 16 + nn], v);
    }
}

// ---------------------------------------------------------------------------
// Node kernel: xn = act(W_n2 @ [x, agg, u[batch]]); also node_sum[batch] += xn
// ---------------------------------------------------------------------------
template <int DN, int DA, int DG, int NO, bool RELU>
__global__ void __launch_bounds__(32) mpnn_node_kernel(
    const float* __restrict__ x, const float* __restrict__ agg,
    const float* __restrict__ u, const int* __restrict__ batch,
    const float* __restrict__ WfN2, const float* __restrict__ bias,
    float* __restrict__ xout, float* __restrict__ node_sum) {
  __shared__ float At[16][AT_STRIDE];
  __shared__ int sB[16];
  const int lane = threadIdx.x;
  const int n0 = blockIdx.x * 16;
  if (lane < 16) sB[lane] = batch[n0 + lane];
  __syncthreads();

  gather_block<DN>(x, n0, At, 0, lane);
  gather_block<DA>(agg, n0, At, DN, lane);
  gather_rows<DG>(u, sB, At, DN + DA, lane);
  wait_async();
  __syncthreads();

  constexpr int K = DN + DA + DG;
  constexpr int NCH = NO / 16;
  v8f acc[NCH];
  zero_acc(acc);
  gemm_tile<K, NCH>(At, WfN2, acc, lane);

  const int hi = lane >> 4, nn = lane & 15;
#pragma unroll
  for (int c = 0; c < NCH; ++c)
#pragma unroll
    for (int r = 0; r < 8; ++r) {
      int mrow = r + 8 * hi;
      float v = acc[c][r] + bias[c * 16 + nn];
      if (RELU) v = fmaxf(v, 0.f);
      xout[(size_t)(n0 + mrow) * NO + c * 16 + nn] = v;
      atomicAddF(&node_sum[(size_t)sB[mrow] * NO + c * 16 + nn], v);
    }
}

// ---------------------------------------------------------------------------
// Global kernel: un = act(W_g @ [node_sum, edge_sum, u]) per graph.
// ---------------------------------------------------------------------------
template <int NO, int EO, int DG, int GO, bool RELU>
__global__ void __launch_bounds__(32) mpnn_global_kernel(
    const float* __restrict__ node_sum, const float* __restrict__ edge_sum,
    const float* __restrict__ u, const float* __restrict__ WfG,
    const float* __restrict__ bias, float* __restrict__ uout) {
  __shared__ float At[16][AT_STRIDE];
  const int lane = threadIdx.x;
  const int g0 = blockIdx.x * 16;

  gather_block<NO>(node_sum, g0, At, 0, lane);
  gather_block<EO>(edge_sum, g0, At, NO, lane);
  gather_block<DG>(u, g0, At, NO + EO, lane);
  wait_async();
  __syncthreads();

  constexpr int K = NO + EO + DG;
  constexpr int NCH = GO / 16;
  v8f acc[NCH];
  zero_acc(acc);
  gemm_tile<K, NCH>(At, WfG, acc, lane);

  const int hi = lane >> 4, nn = lane & 15;
#pragma unroll
  for (int c = 0; c < NCH; ++c)
#pragma unroll
    for (int r = 0; r < 8; ++r) {
      int mrow = r + 8 * hi;
      float v = acc[c][r] + bias[c * 16 + nn];
      if (RELU) v = fmaxf(v, 0.f);
      uout[(size_t)(g0 + mrow) * GO + c * 16 + nn] = v;
    }
}

// W (dout x din, row-major) -> Wf fragment-major: for k-step s = k/4 and
// output chunk c, 64 floats in exact WMMA-B lane order:
//   Wf[((s*NCH + c)*32 + lane)*2 + j] = W[c*16 + (lane&15)][s*4 + ((lane>>4)<<1) + j]
__global__ void pack_wfrag_kernel(const float* __restrict__ W,
                                  float* __restrict__ Wf, int dout, int din) {
  int i = blockIdx.x * blockDim.x + threadIdx.x;
  if (i >= dout * din) return;
  int nch = dout >> 4;
  int j = i & 1;
  int lane = (i >> 1) & 31;
  int rest = i >> 6;
  int c = rest % nch;
  int kstep = rest / nch;
  int k = kstep * 4 + ((lane >> 4) << 1) + j;
  int o = c * 16 + (lane & 15);
  Wf[i] = W[(size_t)o * din + k];
}

// BatchNorm (training mode): per-column sum / sumsq over 64 columns.
__global__ void bn_reduce_kernel(const float* __restrict__ h, int nrows,
                                 float* __restrict__ sums) {
  int col = threadIdx.x & 63;
  int rpb = blockDim.x >> 6;
  int row = blockIdx.x * rpb + (threadIdx.x >> 6);
  int stride = gridDim.x * rpb;
  float s = 0.f, s2 = 0.f;
  for (; row < nrows; row += stride) {
    float v = h[(size_t)row * 64 + col];
    s += v;
    s2 += v * v;
  }
  atomicAddF(&sums[col], s);
  atomicAddF(&sums[64 + col], s2);
}

__global__ void bn_apply_kernel(float* __restrict__ h, int nrows,
                                const float* __restrict__ sums,
                                const float* __restrict__ g,
                                const float* __restrict__ beta) {
  long i = blockIdx.x * (long)blockDim.x + threadIdx.x;
  long total = (long)nrows * 64;
  if (i >= total) return;
  int col = (int)(i & 63);
  float invn = 1.f / (float)nrows;
  float m = sums[col] * invn;
  float var = sums[64 + col] * invn - m * m;
  h[i] = (h[i] - m) * rsqrtf(var + 1e-5f) * g[col] + beta[col];
}

// ---------------------------------------------------------------------------
// Host-side orchestration
// ---------------------------------------------------------------------------
template <int DN_, int DE_, int DG_, int EO_, int NO_, int GO_, bool RELU_>
static void run_meta_layer(const float* xin, const float* ein, const float* uin,
                           const int* src, const int* dst, const int* batch,
                           float* const* wtl, const float* const* Pl,
                           float* eout, float* xout, float* uout, float* agg,
                           float* nsum, float* esum, hipStream_t stream) {
  hipMemsetAsync(agg, 0, (size_t)N_NODES_C * NO_ * sizeof(float), stream);
  hipMemsetAsync(nsum, 0, (size_t)N_GRAPHS_C * NO_ * sizeof(float), stream);
  hipMemsetAsync(esum, 0, (size_t)N_GRAPHS_C * EO_ * sizeof(float), stream);
  mpnn_edge_kernel<DN_, DE_, DG_, EO_, NO_, RELU_>
      <<<N_EDGES_C / 16, 32, 0, stream>>>(xin, ein, uin, src, dst, batch,
                                          wtl[0], Pl[1], wtl[1], Pl[3], eout,
                                          agg, esum);
  mpnn_node_kernel<DN_, NO_, DG_, NO_, RELU_>
      <<<N_NODES_C / 16, 32, 0, stream>>>(xin, agg, uin, batch, wtl[2], Pl[5],
                                          xout, nsum);
  mpnn_global_kernel<NO_, EO_, DG_, GO_, RELU_>
      <<<N_GRAPHS_C / 16, 32, 0, stream>>>(nsum, esum, uin, wtl[3], Pl[7],
                                           uout);
}

static void run_bn(float* h, int nrows, float* sums, const float* g,
                   const float* b, hipStream_t stream) {
  hipMemsetAsync(sums, 0, 128 * sizeof(float), stream);
  int grid = (nrows + 3) / 4;
  if (grid > 2048) grid = 2048;
  bn_reduce_kernel<<<grid, 256, 0, stream>>>(h, nrows, sums);
  long total = (long)nrows * 64;
  bn_apply_kernel<<<(int)((total + 255) / 256), 256, 0, stream>>>(h, nrows,
                                                                  sums, g, b);
}

extern "C" void kernel_launch(void* const* d_in, const int* in_sizes, int n_in,
                              void* d_out, int out_size, void* d_ws,
                              size_t ws_size, hipStream_t stream) {
  (void)in_sizes;
  (void)n_in;
  (void)out_size;
  (void)ws_size;
  const float* x_in = (const float*)d_in[0];
  const int* eidx = (const int*)d_in[1];
  const float* ea_in = (const float*)d_in[2];
  const float* u_in = (const float*)d_in[3];
  const int* batch = (const int*)d_in[4];
  const int* src = eidx;
  const int* dst = eidx + N_EDGES_C;

  const float* P[24];
  for (int i = 0; i < 24; ++i) P[i] = (const float*)d_in[5 + i];
  const float* NM[12];
  for (int i = 0; i < 12; ++i) NM[i] = (const float*)d_in[29 + i];

  const int DN[3] = {32, 64, 64}, DE[3] = {16, 64, 64}, DG[3] = {16, 64, 64};
  const int EO[3] = {64, 64, 32}, NO[3] = {64, 64, 32}, GO[3] = {64, 64, 16};

  // ---- Workspace layout (floats) ----
  float* ws = (float*)d_ws;
  size_t off = 0;
  float* wt[3][4];
  int KSZ[3][4], DSZ[3][4];
  for (int l = 0; l < 3; ++l) {
    KSZ[l][0] = 2 * DN[l] + DE[l] + DG[l];  // edge MLP K
    KSZ[l][1] = 2 * DN[l] + EO[l] + DG[l];  // n1 K
    KSZ[l][2] = DN[l] + NO[l] + DG[l];      // n2 K
    KSZ[l][3] = NO[l] + EO[l] + DG[l];      // global K
    DSZ[l][0] = EO[l];
    DSZ[l][1] = NO[l];
    DSZ[l][2] = NO[l];
    DSZ[l][3] = GO[l];
    for (int m = 0; m < 4; ++m) {
      wt[l][m] = ws + off;
      off += (size_t)DSZ[l][m] * KSZ[l][m];
    }
  }
  float* xbuf = ws + off; off += (size_t)N_NODES_C * 64;
  float* ebuf = ws + off; off += (size_t)N_EDGES_C * 64;
  float* ubuf = ws + off; off += (size_t)N_GRAPHS_C * 64;
  float* agg  = ws + off; off += (size_t)N_NODES_C * 64;
  float* nsum = ws + off; off += (size_t)N_GRAPHS_C * 64;
  float* esum = ws + off; off += (size_t)N_GRAPHS_C * 64;
  float* bns  = ws + off; off += 128;

  // ---- Pre-pack all weight matrices into WMMA-B fragment order ----
  for (int l = 0; l < 3; ++l) {
    const float* wsrc[4] = {P[8 * l + 0], P[8 * l + 2], P[8 * l + 4],
                            P[8 * l + 6]};
    for (int m = 0; m < 4; ++m) {
      int n = DSZ[l][m] * KSZ[l][m];
      pack_wfrag_kernel<<<(n + 255) / 256, 256, 0, stream>>>(wsrc[m], wt[l][m],
                                                             DSZ[l][m],
                                                             KSZ[l][m]);
    }
  }

  float* dxo = (float*)d_out;                       // x out: 20000 x 32
  float* deo = dxo + (size_t)N_NODES_C * 32;        // e out: 320000 x 32
  float* duo = deo + (size_t)N_EDGES_C * 32;        // u out: 128 x 16

  // ---- Layer 0 (in: raw dims, out: HID=64) ----
  run_meta_layer<32, 16, 16, 64, 64, 64, true>(
      x_in, ea_in, u_in, src, dst, batch, wt[0], P, ebuf, xbuf, ubuf, agg,
      nsum, esum, stream);
  run_bn(xbuf, N_NODES_C, bns, NM[0], NM[1], stream);
  run_bn(ebuf, N_EDGES_C, bns, NM[2], NM[3], stream);
  run_bn(ubuf, N_GRAPHS_C, bns, NM[4], NM[5], stream);

  // ---- Layer 1 (64 -> 64, in-place buffers) ----
  run_meta_layer<64, 64, 64, 64, 64, 64, true>(
      xbuf, ebuf, ubuf, src, dst, batch, wt[1], P + 8, ebuf, xbuf, ubuf, agg,
      nsum, esum, stream);
  run_bn(xbuf, N_NODES_C, bns, NM[6], NM[7], stream);
  run_bn(ebuf, N_EDGES_C, bns, NM[8], NM[9], stream);
  run_bn(ubuf, N_GRAPHS_C, bns, NM[10], NM[11], stream);

  // ---- Layer 2 (64 -> outputs, no activation, writes d_out) ----
  run_meta_layer<64, 64, 64, 32, 32, 16, false>(
      xbuf, ebuf, ubuf, src, dst, batch, wt[2], P + 16, deo, dxo, duo, agg,
      nsum, esum, stream);
}